// AdaptedGaussianConditional_7035156431605
// MI455X (gfx1250) — compile-verified
//
#include <hip/hip_runtime.h>

// Elementwise codebook quantize/dequantize (AdaptedGaussianConditional).
// Memory-bound: 16B/element HBM traffic -> ~8.6us floor at 23.3 TB/s.
// Codebook (1KB) staged into LDS via CDNA5 async global->LDS copy, then a
// branchless 8-step binary search per element from LDS.

typedef __attribute__((ext_vector_type(4))) float v4f;
typedef __attribute__((ext_vector_type(4))) int   v4i;

#define NLEV 256

__global__ __launch_bounds__(256) void agc_quant_kernel(
    const float* __restrict__ inputs,
    const float* __restrict__ means,
    const float* __restrict__ codebook,
    float* __restrict__ dequant,
    int*   __restrict__ symbols,
    int n4)  // number of float4 groups
{
    __shared__ float lut[NLEV];

    const unsigned tid = threadIdx.x;

    // ---- Stage the 1KB sorted codebook into LDS with the CDNA5 async
    // global->LDS path (ASYNCcnt-tracked). blockDim.x == 256 == NLEV, so each
    // lane of all 8 waves copies exactly one dword with full EXEC.
    {
        unsigned lds_addr = (unsigned)(size_t)(&lut[0]) + tid * 4u;
        unsigned goff     = tid * 4u;
        asm volatile(
            "global_load_async_to_lds_b32 %0, %1, %2\n\t"
            "s_wait_asynccnt 0x0"
            :
            : "v"(lds_addr), "v"(goff), "s"(codebook)
            : "memory");
    }
    __syncthreads();

    long gid = (long)blockIdx.x * 256 + tid;
    if (gid >= n4) return;

    // Streaming, no reuse -> non-temporal 128-bit loads/stores.
    v4f x = __builtin_nontemporal_load((const v4f*)inputs + gid);
    v4f m = __builtin_nontemporal_load((const v4f*)means  + gid);

    v4f dq;
    v4i sy;
    #pragma unroll
    for (int c = 0; c < 4; ++c) {
        float v = x[c] - m[c];

        // Branchless searchsorted(side='left') over 256 sorted values:
        // pos = min(#{ lut[i] < v }, 255)
        int pos = 0;
        #pragma unroll
        for (int step = 128; step >= 1; step >>= 1) {
            float probe = lut[pos + step - 1];
            pos += (probe < v) ? step : 0;
        }
        int idx = (pos < 1) ? 1 : pos;      // == clip(searchsorted, 1, 255)

        float left  = lut[idx - 1];
        float right = lut[idx];
        // Ties pick the lower index (argmin semantics) -> use <=
        bool pickL = __builtin_fabsf(v - left) <= __builtin_fabsf(v - right);
        int   s = pickL ? (idx - 1) : idx;
        float q = pickL ? left : right;

        sy[c] = s;
        dq[c] = q + m[c];
    }

    __builtin_nontemporal_store(dq, (v4f*)dequant + gid);
    __builtin_nontemporal_store(sy, (v4i*)symbols + gid);
}

extern "C" void kernel_launch(void* const* d_in, const int* in_sizes, int n_in,
                              void* d_out, int out_size, void* d_ws, size_t ws_size,
                              hipStream_t stream) {
    // Inputs per reference setup_inputs(): inputs, means, unique_values (all f32)
    const float* inputs = (const float*)d_in[0];
    const float* means  = (const float*)d_in[1];
    const float* uv     = (const float*)d_in[2];

    const long n = (long)in_sizes[0];      // 16*192*64*64 = 12,582,912 (divisible by 4)

    // Tuple output: dequant (n float32) then symbols (n int32), concatenated flat.
    float* dq  = (float*)d_out;
    int*   sym = (int*)(dq + n);

    int n4 = (int)(n / 4);
    int blocks = (n4 + 255) / 256;
    agc_quant_kernel<<<blocks, 256, 0, stream>>>(inputs, means, uv, dq, sym, n4);
}